// XFeat_781684047936
// MI455X (gfx1250) — compile-verified
//
#include <hip/hip_runtime.h>
#include <stdint.h>

// ---------------- problem constants ----------------
#define B_    32
#define H8_   60
#define W8_   88
#define HF_   480
#define WF_   704
#define NPIX  (HF_*WF_)        // 337920
#define NC_   (H8_*W8_)        // 5280
#define NCELL (B_*NC_)         // 168960
#define TOPK_ 4096
#define THR_  0.05f
#define EPS_  1e-12f
#define CAP_  8192             // per-batch peak capacity (LDS sort size, 64KB keys)
#define WORDS_ (NPIX/32)       // 10560 bitmap words per batch

// ---------------- CDNA5 async-to-LDS (probe-confirmed on this toolchain) --------------
#if defined(__AMDGCN__) && __has_builtin(__builtin_amdgcn_global_load_async_to_lds_b128)
#define ASYNC_LDS 1
typedef int v4i_ __attribute__((ext_vector_type(4)));
typedef __attribute__((address_space(1))) v4i_* as1_v4ip;
typedef __attribute__((address_space(3))) v4i_* as3_v4ip;
__device__ __forceinline__ void async_wait_all() {
#if __has_builtin(__builtin_amdgcn_s_wait_asynccnt)
  __builtin_amdgcn_s_wait_asynccnt(0);
#else
  asm volatile("s_wait_asynccnt 0" ::: "memory");
#endif
}
#else
#define ASYNC_LDS 0
#endif

// ---------------- CDNA5 WMMA f32 16x16x4 ----------------
#if defined(__AMDGCN__) && __has_builtin(__builtin_amdgcn_wmma_f32_16x16x4_f32)
#define HAVE_WMMA4 1
typedef float v2f_ __attribute__((ext_vector_type(2)));
typedef float v8f_ __attribute__((ext_vector_type(8)));
#endif

// ---------------- kernel 1: softmax(K1,65ch) -> pixel-shuffled heatmap ----------------
__global__ __launch_bounds__(256) void k_softmax_hm(const float* __restrict__ K1,
                                                    float* __restrict__ hm) {
  int t = blockIdx.x * 256 + threadIdx.x;
  if (t >= NCELL) return;
  int b = t / NC_, cell = t - b * NC_;
  int y8 = cell / W8_, x8 = cell - y8 * W8_;
  const float* kp = K1 + (size_t)b * 65 * NC_ + cell;
  float v[65];
  float m = -3.4e38f;
#pragma unroll
  for (int c = 0; c < 65; c++) { v[c] = kp[(size_t)c * NC_]; m = fmaxf(m, v[c]); }
  float s = 0.f;
#pragma unroll
  for (int c = 0; c < 65; c++) { v[c] = __expf(v[c] - m); s += v[c]; }
  float inv = 1.0f / s;
  // hm[b, y8*8 + r, x8*8 + q] = softmax[c = 8*r + q]
  float* hb = hm + (size_t)b * NPIX + (size_t)(y8 * 8) * WF_ + x8 * 8;
#pragma unroll
  for (int r = 0; r < 8; r++) {
    float4 lo = make_float4(v[8*r+0]*inv, v[8*r+1]*inv, v[8*r+2]*inv, v[8*r+3]*inv);
    float4 hi = make_float4(v[8*r+4]*inv, v[8*r+5]*inv, v[8*r+6]*inv, v[8*r+7]*inv);
    float* row = hb + (size_t)r * WF_;
    ((float4*)row)[0] = lo;
    ((float4*)row)[1] = hi;
  }
}

// ---------------- kernel 2: channel-normalize M1 + transpose to channels-last ---------
__global__ __launch_bounds__(256) void k_norm_tr(const float* __restrict__ M1,
                                                 float* __restrict__ M1t) {
  int t = blockIdx.x * 256 + threadIdx.x;
  if (t >= NCELL) return;
  int b = t / NC_, cell = t - b * NC_;
  const float* mp = M1 + (size_t)b * 64 * NC_ + cell;
  float v[64];
  float ss = 0.f;
#pragma unroll
  for (int c = 0; c < 64; c++) { float x = mp[(size_t)c * NC_]; v[c] = x; ss += x * x; }
  float inv = 1.0f / fmaxf(sqrtf(ss), EPS_);
  float4* op = (float4*)(M1t + (size_t)t * 64);
#pragma unroll
  for (int q = 0; q < 16; q++)
    op[q] = make_float4(v[4*q]*inv, v[4*q+1]*inv, v[4*q+2]*inv, v[4*q+3]*inv);
}

// ---------------- kernel 3: 5x5 NMS + score, append peaks ----------------
#define TW_ 64
#define TH_ 4
#define LW_ 80   // LDS tile width  (x0-8 .. x0+72), 16B aligned
#define LH_ 8    // LDS tile height (y0-2 .. y0+6)
__global__ __launch_bounds__(256) void k_nms_score(const float* __restrict__ hm,
                                                   const float* __restrict__ H1,
                                                   float* __restrict__ cs,
                                                   int* __restrict__ ci,
                                                   int* __restrict__ cnt,
                                                   unsigned* __restrict__ bitmap) {
  __shared__ __align__(16) float tile[LH_ * LW_];
  int b = blockIdx.z;
  int tx0 = blockIdx.x * TW_, ty0 = blockIdx.y * TH_;
  int xl = tx0 - 8, yl = ty0 - 2;
  const float* hb = hm + (size_t)b * NPIX;
  int t = threadIdx.x;
  bool interior = (xl >= 0) && (xl + LW_ <= WF_) && (yl >= 0) && (yl + LH_ <= HF_);
  if (interior) {
#if ASYNC_LDS
    if (t < (LH_ * LW_ / 4)) {                 // 160 b128 async transfers
      int row = t / (LW_ / 4), c4 = t - row * (LW_ / 4);
      const float* g = hb + (size_t)(yl + row) * WF_ + xl + c4 * 4;
      float* l = &tile[row * LW_ + c4 * 4];
      __builtin_amdgcn_global_load_async_to_lds_b128(
          (as1_v4ip)const_cast<float*>(g), (as3_v4ip)l, 0, 0);
    }
    async_wait_all();
#else
    for (int i = t; i < LH_ * LW_ / 4; i += 256) {
      int row = i / (LW_ / 4), c4 = i - row * (LW_ / 4);
      float4 d = *(const float4*)(hb + (size_t)(yl + row) * WF_ + xl + c4 * 4);
      *(float4*)&tile[row * LW_ + c4 * 4] = d;
    }
#endif
  } else {
    for (int i = t; i < LH_ * LW_; i += 256) {
      int row = i / LW_, col = i - row * LW_;
      int gy = yl + row, gx = xl + col;
      tile[i] = (gx >= 0 && gx < WF_ && gy >= 0 && gy < HF_)
                  ? hb[(size_t)gy * WF_ + gx] : -INFINITY;
    }
  }
  __syncthreads();

  int lx = t & (TW_ - 1), ly = t / TW_;
  int x = tx0 + lx, y = ty0 + ly;
  if (x >= WF_ || y >= HF_) return;
  float v = tile[(ly + 2) * LW_ + 8 + lx];
  float mx = -INFINITY;
#pragma unroll
  for (int dy = 0; dy < 5; dy++)
#pragma unroll
    for (int dx = 0; dx < 5; dx++)
      mx = fmaxf(mx, tile[(ly + dy) * LW_ + (6 + lx + dx)]);
  int idx = y * WF_ + x;
  if (!((v == mx) && (v > THR_)) || idx == 0) return;

  // s_k: nearest resample of hm at identity-ish grid (align_corners mapping)
  float gy_ = 2.0f * (float)y / (float)(HF_ - 1) - 1.0f;
  float gx_ = 2.0f * (float)x / (float)(WF_ - 1) - 1.0f;
  float fiy = ((gy_ + 1.0f) * (float)HF_ - 1.0f) * 0.5f;
  float fix = ((gx_ + 1.0f) * (float)WF_ - 1.0f) * 0.5f;
  int niy = (int)rintf(fiy), nix = (int)rintf(fix);
  float sk = 0.f;
  if (niy >= 0 && niy < HF_ && nix >= 0 && nix < WF_) sk = hb[(size_t)niy * WF_ + nix];

  // s_r: bilinear on H1 (60x88)
  float riy = ((gy_ + 1.0f) * (float)H8_ - 1.0f) * 0.5f;
  float rix = ((gx_ + 1.0f) * (float)W8_ - 1.0f) * 0.5f;
  float fy0 = floorf(riy), fx0 = floorf(rix);
  float tyf = riy - fy0, txf = rix - fx0;
  int iy0 = (int)fy0, ix0 = (int)fx0;
  const float* Hb = H1 + (size_t)b * NC_;
  float g4[4];
#pragma unroll
  for (int q = 0; q < 4; q++) {
    int yy = iy0 + (q >> 1), xx = ix0 + (q & 1);
    bool ok = (yy >= 0 && yy < H8_ && xx >= 0 && xx < W8_);
    int yc = min(max(yy, 0), H8_ - 1), xc = min(max(xx, 0), W8_ - 1);
    g4[q] = ok ? Hb[yc * W8_ + xc] : 0.f;
  }
  float top = g4[0] * (1.f - txf) + g4[1] * txf;
  float bot = g4[2] * (1.f - txf) + g4[3] * txf;
  float sr  = top * (1.f - tyf) + bot * tyf;

  float score = sk * sr;
  int slot = atomicAdd(&cnt[b], 1);
  if (slot < CAP_) { cs[b * CAP_ + slot] = score; ci[b * CAP_ + slot] = idx; }
  atomicOr(&bitmap[(size_t)b * WORDS_ + (idx >> 5)], 1u << (idx & 31));
}

// ---------------- kernel 4: per-batch exact top-k (sort peaks, -1 fill by index) -------
__global__ __launch_bounds__(1024) void k_topk(const float* __restrict__ cs,
                                               const int* __restrict__ ci,
                                               const int* __restrict__ cnt,
                                               const unsigned* __restrict__ bitmap,
                                               float* __restrict__ mkpts,
                                               float* __restrict__ vals) {
  __shared__ unsigned long long key[CAP_];   // 64KB
  int b = blockIdx.x, t = threadIdx.x;
  int P = min(cnt[b], CAP_);
  for (int i = t; i < CAP_; i += 1024) {
    unsigned long long k;
    if (i < P) {
      unsigned u = __float_as_uint(cs[b * CAP_ + i]);
      unsigned o = (u & 0x80000000u) ? ~u : (u | 0x80000000u);   // order-preserving
      k = (((unsigned long long)(~o)) << 32) | (unsigned)ci[b * CAP_ + i];
    } else k = ~0ull;                                            // sentinel sorts last
    key[i] = k;
  }
  __syncthreads();
  // bitonic sort ascending -> (score desc, idx asc)
  for (int kk = 2; kk <= CAP_; kk <<= 1) {
    for (int j = kk >> 1; j > 0; j >>= 1) {
      for (int i = t; i < CAP_; i += 1024) {
        int ixj = i ^ j;
        if (ixj > i) {
          bool up = ((i & kk) == 0);
          unsigned long long a = key[i], c = key[ixj];
          if ((a > c) == up) { key[i] = c; key[ixj] = a; }
        }
      }
      __syncthreads();
    }
  }
  int Peff = min(P, TOPK_);
  for (int s = t; s < Peff; s += 1024) {
    unsigned long long k = key[s];
    unsigned o = ~(unsigned)(k >> 32);
    unsigned u = (o & 0x80000000u) ? (o & 0x7FFFFFFFu) : ~o;
    int idx = (int)(unsigned)(k & 0xFFFFFFFFu);
    vals[b * TOPK_ + s] = __uint_as_float(u);
    mkpts[(size_t)(b * TOPK_ + s) * 2 + 0] = (float)(idx % WF_);
    mkpts[(size_t)(b * TOPK_ + s) * 2 + 1] = (float)(idx / WF_);
  }
  __syncthreads();
  int need = TOPK_ - Peff;
  if (need <= 0) return;
  // fill with smallest non-peak indices (value -1), matching top_k tie order
  int* sc = (int*)key;                      // reuse LDS: ping-pong scan buffers
  const unsigned* bm = bitmap + (size_t)b * WORDS_;
  int base = 0;
  for (int chunk = 0; chunk * 1024 < WORDS_ && base < need; chunk++) {
    int w = chunk * 1024 + t;
    unsigned word = (w < WORDS_) ? bm[w] : 0xFFFFFFFFu;
    int zc = 32 - __popc(word);
    sc[t] = zc;
    __syncthreads();
    int rb = 0, wb = 1024;
    for (int off = 1; off < 1024; off <<= 1) {
      int v2 = sc[rb + t];
      if (t >= off) v2 += sc[rb + t - off];
      sc[wb + t] = v2;
      __syncthreads();
      int tmp = rb; rb = wb; wb = tmp;
    }
    int incl = sc[rb + t];
    int total = sc[rb + 1023];
    int rank = base + (incl - zc);
    if (zc > 0 && rank < need) {
      for (int bit = 0; bit < 32 && rank < need; bit++) {
        if (!((word >> bit) & 1)) {
          int idx = w * 32 + bit;
          int s2 = Peff + rank;
          vals[b * TOPK_ + s2] = -1.0f;
          mkpts[(size_t)(b * TOPK_ + s2) * 2 + 0] = (float)(idx % WF_);
          mkpts[(size_t)(b * TOPK_ + s2) * 2 + 1] = (float)(idx / WF_);
          rank++;
        }
      }
    }
    base += total;
    __syncthreads();
  }
}

// ---------------- kernel 5: bicubic descriptor gather + L2 normalize ----------------
__device__ __forceinline__ float cubw(float x) {   // a = -0.75
  x = fabsf(x);
  if (x <= 1.f) return (1.25f * x - 2.25f) * x * x + 1.f;
  if (x <  2.f) return -0.75f * ((((x - 5.f) * x) + 8.f) * x - 4.f);
  return 0.f;
}

// One wave32 per keypoint. The 16-tap x 64-channel bicubic blend is run on the
// matrix pipe: D(16x16) = A(16x4) x B(4x16) + C, chained over the 4 tap-rows.
// A holds the 4 tap weights of row j replicated across all 16 M-rows
// (lanes 0-15 carry taps i={0,1}, lanes 16-31 taps i={2,3}, per ISA A 16x4 layout),
// B holds the gathered channel values (lane&15 = channel within 16-ch group).
// All D rows are identical copies of the true 1x16 result; element 0 is used.
__global__ __launch_bounds__(256) void k_feats(const float* __restrict__ M1t,
                                               const float* __restrict__ mkpts,
                                               float* __restrict__ feats) {
  int wid = threadIdx.x >> 5, lane = threadIdx.x & 31;
  int kp = blockIdx.x * 8 + wid;                    // B_*TOPK_ keypoints
  int b = kp >> 12;
  float mx = mkpts[(size_t)kp * 2 + 0];
  float my = mkpts[(size_t)kp * 2 + 1];
  float gx = 2.0f * mx / (float)(WF_ - 1) - 1.0f;
  float gy = 2.0f * my / (float)(HF_ - 1) - 1.0f;
  float fix = ((gx + 1.0f) * (float)W8_ - 1.0f) * 0.5f;
  float fiy = ((gy + 1.0f) * (float)H8_ - 1.0f) * 0.5f;
  float fx0 = floorf(fix), fy0 = floorf(fiy);
  float tx = fix - fx0, ty = fiy - fy0;
  int ix0 = (int)fx0, iy0 = (int)fy0;
  float wx[4] = { cubw(tx + 1.f), cubw(tx), cubw(1.f - tx), cubw(2.f - tx) };
  float wy[4] = { cubw(ty + 1.f), cubw(ty), cubw(1.f - ty), cubw(2.f - ty) };
  const float* Mb = M1t + (size_t)b * NC_ * 64;
  float* op = feats + (size_t)kp * 64;

#ifdef HAVE_WMMA4
  int half = lane >> 4;                 // 0 -> taps i={0,1}; 1 -> taps i={2,3}
  int n = lane & 15;                    // channel within 16-channel group
  int i0 = 2 * half, i1 = i0 + 1;
  int xx0 = ix0 + i0 - 1, xx1 = ix0 + i1 - 1;
  bool vx0 = (xx0 >= 0 && xx0 < W8_), vx1 = (xx1 >= 0 && xx1 < W8_);
  int xc0 = min(max(xx0, 0), W8_ - 1), xc1 = min(max(xx1, 0), W8_ - 1);
  v8f_ acc[4];
#pragma unroll
  for (int cg = 0; cg < 4; cg++) { v8f_ z = {0.f,0.f,0.f,0.f,0.f,0.f,0.f,0.f}; acc[cg] = z; }
#pragma unroll
  for (int g = 0; g < 4; g++) {         // tap row j == g
    int yy = iy0 + g - 1;
    bool vy = (yy >= 0 && yy < H8_);
    int yc = min(max(yy, 0), H8_ - 1);
    const float* r0 = Mb + ((size_t)yc * W8_ + xc0) * 64;
    const float* r1 = Mb + ((size_t)yc * W8_ + xc1) * 64;
    v2f_ a; a[0] = wy[g] * wx[i0]; a[1] = wy[g] * wx[i1];
#pragma unroll
    for (int cg = 0; cg < 4; cg++) {
      int ch = cg * 16 + n;
      v2f_ bv;
      bv[0] = (vy && vx0) ? r0[ch] : 0.f;
      bv[1] = (vy && vx1) ? r1[ch] : 0.f;
      acc[cg] = __builtin_amdgcn_wmma_f32_16x16x4_f32(
          false, a, false, bv, (short)0, acc[cg], false, false);
    }
  }
  float s = 0.f;
#pragma unroll
  for (int cg = 0; cg < 4; cg++) s += acc[cg][0] * acc[cg][0];
#pragma unroll
  for (int m = 16; m > 0; m >>= 1) s += __shfl_xor(s, m, 32);
  s *= 0.5f;                            // every channel counted by both wave halves
  float inv = 1.0f / fmaxf(sqrtf(s), EPS_);
  float e0 = half ? acc[2][0] : acc[0][0];
  float e1 = half ? acc[3][0] : acc[1][0];
  op[half * 32 + n]      = e0 * inv;
  op[half * 32 + 16 + n] = e1 * inv;
#else
  float a0 = 0.f, a1 = 0.f;
#pragma unroll
  for (int j = 0; j < 4; j++) {
    int yy = iy0 + j - 1;
    bool vy = (yy >= 0 && yy < H8_);
    int yc = min(max(yy, 0), H8_ - 1);
#pragma unroll
    for (int i = 0; i < 4; i++) {
      int xx = ix0 + i - 1;
      bool ok = vy && (xx >= 0) && (xx < W8_);
      int xc = min(max(xx, 0), W8_ - 1);
      float w = wy[j] * wx[i];
      const float* p = Mb + ((size_t)yc * W8_ + xc) * 64;
      float v0 = p[lane], v1 = p[lane + 32];
      if (!ok) { v0 = 0.f; v1 = 0.f; }
      a0 = fmaf(w, v0, a0);
      a1 = fmaf(w, v1, a1);
    }
  }
  float s = a0 * a0 + a1 * a1;
#pragma unroll
  for (int m = 16; m > 0; m >>= 1) s += __shfl_xor(s, m, 32);
  float inv = 1.0f / fmaxf(sqrtf(s), EPS_);
  op[lane] = a0 * inv;
  op[lane + 32] = a1 * inv;
#endif
}

// ---------------- launch ----------------
extern "C" void kernel_launch(void* const* d_in, const int* in_sizes, int n_in,
                              void* d_out, int out_size, void* d_ws, size_t ws_size,
                              hipStream_t stream) {
  const float* M1 = (const float*)d_in[0];   // (32,64,60,88)
  const float* K1 = (const float*)d_in[1];   // (32,65,60,88)
  const float* H1 = (const float*)d_in[2];   // (32,1,60,88)
  float* out = (float*)d_out;
  float* mkpts = out;                          // (32,4096,2)
  float* vals  = out + (size_t)B_ * TOPK_ * 2; // (32,4096)
  float* feats = out + (size_t)B_ * TOPK_ * 3; // (32,4096,64)

  float* ws   = (float*)d_ws;
  float* hm   = ws;                                   // B*NPIX
  float* M1t  = hm  + (size_t)B_ * NPIX;              // B*NPIX (== B*NC*64)
  float* cs   = M1t + (size_t)B_ * NPIX;              // B*CAP
  int*   ci   = (int*)(cs + (size_t)B_ * CAP_);       // B*CAP
  int*   cnt  = (int*)(ci + (size_t)B_ * CAP_);       // B
  unsigned* bitmap = (unsigned*)(cnt + B_);           // B*WORDS

  // zero peak counters + bitmap (contiguous region)
  (void)hipMemsetAsync(cnt, 0, (size_t)(B_ + B_ * WORDS_) * sizeof(int), stream);

  k_softmax_hm<<<NCELL / 256, 256, 0, stream>>>(K1, hm);
  k_norm_tr  <<<NCELL / 256, 256, 0, stream>>>(M1, M1t);
  dim3 gC(WF_ / TW_, HF_ / TH_, B_);
  k_nms_score<<<gC, 256, 0, stream>>>(hm, H1, cs, ci, cnt, bitmap);
  k_topk<<<B_, 1024, 0, stream>>>(cs, ci, cnt, bitmap, mkpts, vals);
  k_feats<<<(B_ * TOPK_) / 8, 256, 0, stream>>>(M1t, mkpts, feats);
}